// SAGE_89739046682851
// MI455X (gfx1250) — compile-verified
//
#include <hip/hip_runtime.h>
#include <hip/hip_bf16.h>

// ---------------- problem constants (from reference) ----------------
#define N0 600000
#define N1 60000
#define N2 6000
#define N3 1000
#define E0 600000
#define E1 60000
#define E2 10000
#define D_IN 256
#define D_H 256
#define N_CLS 47

typedef __attribute__((ext_vector_type(2))) float v2f;
typedef __attribute__((ext_vector_type(8))) float v8f;

// ---------------- utility kernels ----------------
__global__ void sage_zero_kernel(float* __restrict__ p, long long n) {
    long long stride = (long long)gridDim.x * blockDim.x;
    for (long long i = (long long)blockIdx.x * blockDim.x + threadIdx.x; i < n; i += stride)
        p[i] = 0.0f;
}

// One edge handled by 64 threads; each thread moves one float4 (16B) of the
// 256-float row: fully coalesced 1KB gather per edge, 4 f32 atomics per thread.
__global__ void sage_edge_scatter_kernel(const float* __restrict__ h,
                                         const int* __restrict__ src,
                                         const int* __restrict__ dst,
                                         int n_edges,
                                         float* __restrict__ agg,
                                         float* __restrict__ deg) {
    unsigned t = blockIdx.x * blockDim.x + threadIdx.x;
    unsigned e = t >> 6;         // edge index
    unsigned c = t & 63;         // float4 chunk within row
    if (e >= (unsigned)n_edges) return;
    int s = src[e];
    int d = dst[e];
    const float4 v = ((const float4*)(h + (size_t)s * D_IN))[c];
    float* out = agg + (size_t)d * D_IN + (size_t)c * 4;
    unsafeAtomicAdd(out + 0, v.x);
    unsafeAtomicAdd(out + 1, v.y);
    unsafeAtomicAdd(out + 2, v.z);
    unsafeAtomicAdd(out + 3, v.w);
    if (c == 0) unsafeAtomicAdd(deg + d, 1.0f);
}

// agg[n][:] *= 1 / max(deg[n], 1)
__global__ void sage_normalize_kernel(float* __restrict__ agg,
                                      const float* __restrict__ deg,
                                      int n_rows) {
    unsigned t = blockIdx.x * blockDim.x + threadIdx.x;
    unsigned n = t >> 6;
    unsigned c = t & 63;
    if (n >= (unsigned)n_rows) return;
    float inv = 1.0f / fmaxf(deg[n], 1.0f);
    float4* p = (float4*)(agg + (size_t)n * D_IN) + c;
    float4 v = *p;
    v.x *= inv; v.y *= inv; v.z *= inv; v.w *= inv;
    *p = v;
}

// ---------------- WMMA lane layouts (V_WMMA_F32_16X16X4_F32) ----------------
// A (16x4 f32): lanes 0-15 hold row M=lane&15, VGPR0=K(k+0),VGPR1=K(k+1);
//               lanes 16-31 hold K(k+2),K(k+3).
// B (4x16 f32): VGPR0 lanes0-15 = row K(k+0), lanes16-31 = row K(k+2);
//               VGPR1 = rows K(k+1)/K(k+3). (columns = lane&15)
// C/D (16x16) : VGPR i, lanes0-15 -> M=i, lanes16-31 -> M=i+8.

// One K-sweep (K=256) of A(16x256) x W(256x64-col-strip) into 4 accumulators.
// Branch-free: pointer-bump addressing, compile-time W stride (256).
__device__ __forceinline__ void sage_pass256(v8f acc[4],
                                             const float* __restrict__ in,
                                             const float* __restrict__ W,
                                             int row, int colbase, int half) {
    const float* ap = in + (size_t)row * D_IN + (half << 1);
    const float* wp = W + (size_t)(half << 1) * D_H + colbase;
#pragma unroll 2
    for (int k = 0; k < D_IN; k += 4) {
        v2f a = *(const v2f*)ap;           // global_load_b64, 8B aligned
        ap += 4;
#pragma unroll
        for (int j = 0; j < 4; ++j) {
            v2f b;
            b.x = wp[16 * j];
            b.y = wp[16 * j + D_H];        // constant 1024B immediate offset
            acc[j] = __builtin_amdgcn_wmma_f32_16x16x4_f32(
                false, a, false, b, (short)0, acc[j], false, false);
        }
        wp += 4 * D_H;
    }
}

// Full layers (n_rows % 16 == 0, d_out == 256, ReLU): one wave computes a
// 16-row x 64-col output strip with 4 v8f accumulators. No masking anywhere.
__global__ __launch_bounds__(256)
void sage_gemm256_kernel(const float* __restrict__ self_in,
                         const float* __restrict__ agg_in,
                         const float* __restrict__ Wself,
                         const float* __restrict__ Wneigh,
                         const float* __restrict__ bias,
                         float* __restrict__ out,
                         int n_rows) {
    const int lane = threadIdx.x & 31;
    const int wave = threadIdx.x >> 5;
    const int tile = blockIdx.x * (blockDim.x >> 5) + wave;
    const int tiles_n = D_H / 64;                       // 4 col-groups of 64
    if (tile >= (n_rows >> 4) * tiles_n) return;        // wave-uniform

    const int tm = tile / tiles_n;
    const int tg = tile % tiles_n;
    const int half = lane >> 4;
    const int lm = lane & 15;

    const int row = tm * 16 + lm;
    const int colbase = tg * 64 + lm;

    v8f acc[4];
#pragma unroll
    for (int j = 0; j < 4; ++j) acc[j] = (v8f){0.f,0.f,0.f,0.f,0.f,0.f,0.f,0.f};

    sage_pass256(acc, self_in, Wself, row, colbase, half);
    sage_pass256(acc, agg_in, Wneigh, row, colbase, half);

#pragma unroll
    for (int j = 0; j < 4; ++j) {
        const int col = colbase + 16 * j;
        const float bj = bias[col];
#pragma unroll
        for (int i = 0; i < 8; ++i) {
            const int r = tm * 16 + i + 8 * half;
            out[(size_t)r * D_H + col] = fmaxf(acc[j][i] + bj, 0.0f);
        }
    }
}

// Tail layer (1000 rows x 47 cols, no ReLU): 16x16 tile per wave. Loads use
// CLAMPED indices (always in-bounds); out-of-range A rows / B cols only feed
// output rows/cols that are masked at the store, so no operand zeroing and no
// exec-mask churn in the inner loop.
__global__ __launch_bounds__(256)
void sage_gemm_tail_kernel(const float* __restrict__ self_in,
                           const float* __restrict__ agg_in,
                           const float* __restrict__ Wself,
                           const float* __restrict__ Wneigh,
                           const float* __restrict__ bias,
                           float* __restrict__ out,
                           int n_rows, int d_out) {
    const int lane = threadIdx.x & 31;
    const int wave = threadIdx.x >> 5;
    const int tiles_n = (d_out + 15) >> 4;
    const int tiles_m = (n_rows + 15) >> 4;
    const int tile = blockIdx.x * (blockDim.x >> 5) + wave;
    if (tile >= tiles_m * tiles_n) return;              // wave-uniform

    const int tm = tile / tiles_n;
    const int tn = tile % tiles_n;
    const int half = lane >> 4;
    const int lm = lane & 15;

    const int row = tm * 16 + lm;
    const int col = tn * 16 + lm;
    const int rr = row < n_rows ? row : n_rows - 1;     // clamp: loads stay legal
    const int cc = col < d_out ? col : d_out - 1;
    const bool col_ok = col < d_out;

    v8f acc = {0.f,0.f,0.f,0.f,0.f,0.f,0.f,0.f};

    const float* ap0 = self_in + (size_t)rr * D_IN + (half << 1);
    const float* ap1 = agg_in + (size_t)rr * D_IN + (half << 1);
    const float* wp0 = Wself + (size_t)(half << 1) * d_out + cc;
    const float* wp1 = Wneigh + (size_t)(half << 1) * d_out + cc;

#pragma unroll 4
    for (int k = 0; k < D_IN; k += 4) {
        v2f a = *(const v2f*)ap0;  ap0 += 4;
        v2f b;  b.x = wp0[0];  b.y = wp0[d_out];  wp0 += 4 * d_out;
        acc = __builtin_amdgcn_wmma_f32_16x16x4_f32(false, a, false, b, (short)0, acc, false, false);
    }
#pragma unroll 4
    for (int k = 0; k < D_IN; k += 4) {
        v2f a = *(const v2f*)ap1;  ap1 += 4;
        v2f b;  b.x = wp1[0];  b.y = wp1[d_out];  wp1 += 4 * d_out;
        acc = __builtin_amdgcn_wmma_f32_16x16x4_f32(false, a, false, b, (short)0, acc, false, false);
    }

    const float bj = bias[cc];
#pragma unroll
    for (int i = 0; i < 8; ++i) {
        const int r = tm * 16 + i + 8 * half;
        if (r < n_rows && col_ok)
            out[(size_t)r * d_out + col] = acc[i] + bj;
    }
}

// ---------------- host-side orchestration ----------------
static inline int ceil_div(long long a, long long b) { return (int)((a + b - 1) / b); }

extern "C" void kernel_launch(void* const* d_in, const int* in_sizes, int n_in,
                              void* d_out, int out_size, void* d_ws, size_t ws_size,
                              hipStream_t stream) {
    const float* x    = (const float*)d_in[0];
    const int*   src0 = (const int*)d_in[1];
    const int*   dst0 = (const int*)d_in[2];
    const int*   src1 = (const int*)d_in[3];
    const int*   dst1 = (const int*)d_in[4];
    const int*   src2 = (const int*)d_in[5];
    const int*   dst2 = (const int*)d_in[6];
    const float* Ws0  = (const float*)d_in[7];
    const float* Wn0  = (const float*)d_in[8];
    const float* b0   = (const float*)d_in[9];
    const float* Ws1  = (const float*)d_in[10];
    const float* Wn1  = (const float*)d_in[11];
    const float* b1   = (const float*)d_in[12];
    const float* Ws2  = (const float*)d_in[13];
    const float* Wn2  = (const float*)d_in[14];
    const float* b2   = (const float*)d_in[15];

    // Workspace layout (floats):
    //   [0, 15.36M)        h0  (N1 x 256)
    //   [15.36M, 30.72M)   scratch region: agg0 (whole) -> later agg1 | h1 | agg2
    //   [30.72M, +70000)   deg (reused per layer)
    float* ws   = (float*)d_ws;
    float* h0   = ws;
    float* scr  = ws + 15360000ll;
    float* deg  = ws + 30720000ll;
    float* agg0 = scr;                       // N1*256
    float* agg1 = scr;                       // N2*256 (agg0 dead by then)
    float* h1   = scr + 1536000ll;           // N2*256
    float* agg2 = scr + 3072000ll;           // N3*256
    float* hout = (float*)d_out;             // N3 x 47

    const int TB = 256;

    // ---------------- layer 0 ----------------
    sage_zero_kernel<<<4096, TB, 0, stream>>>(agg0, (long long)N1 * D_IN);
    sage_zero_kernel<<<64, TB, 0, stream>>>(deg, (long long)N1);
    sage_edge_scatter_kernel<<<ceil_div((long long)E0 * 64, TB), TB, 0, stream>>>(
        x, src0, dst0, E0, agg0, deg);
    sage_normalize_kernel<<<ceil_div((long long)N1 * 64, TB), TB, 0, stream>>>(agg0, deg, N1);
    {
        int tiles = (N1 / 16) * (D_H / 64);
        sage_gemm256_kernel<<<ceil_div(tiles, 8), TB, 0, stream>>>(
            x, agg0, Ws0, Wn0, b0, h0, N1);
    }

    // ---------------- layer 1 ----------------
    sage_zero_kernel<<<512, TB, 0, stream>>>(agg1, (long long)N2 * D_IN);
    sage_zero_kernel<<<8, TB, 0, stream>>>(deg, (long long)N2);
    sage_edge_scatter_kernel<<<ceil_div((long long)E1 * 64, TB), TB, 0, stream>>>(
        h0, src1, dst1, E1, agg1, deg);
    sage_normalize_kernel<<<ceil_div((long long)N2 * 64, TB), TB, 0, stream>>>(agg1, deg, N2);
    {
        int tiles = (N2 / 16) * (D_H / 64);
        sage_gemm256_kernel<<<ceil_div(tiles, 8), TB, 0, stream>>>(
            h0, agg1, Ws1, Wn1, b1, h1, N2);
    }

    // ---------------- layer 2 (tail: 1000 x 47, no ReLU) ----------------
    sage_zero_kernel<<<128, TB, 0, stream>>>(agg2, (long long)N3 * D_IN);
    sage_zero_kernel<<<2, TB, 0, stream>>>(deg, (long long)N3);
    sage_edge_scatter_kernel<<<ceil_div((long long)E2 * 64, TB), TB, 0, stream>>>(
        h1, src2, dst2, E2, agg2, deg);
    sage_normalize_kernel<<<ceil_div((long long)N3 * 64, TB), TB, 0, stream>>>(agg2, deg, N3);
    {
        int tiles = ((N3 + 15) / 16) * ((N_CLS + 15) / 16);
        sage_gemm_tail_kernel<<<ceil_div(tiles, 8), TB, 0, stream>>>(
            h1, agg2, Ws2, Wn2, b2, hout, N3, N_CLS);
    }
}